// ImaginationMamba_15083925143692
// MI455X (gfx1250) — compile-verified
//
#include <hip/hip_runtime.h>
#include <cmath>

typedef __attribute__((ext_vector_type(2))) float v2f;
typedef __attribute__((ext_vector_type(8))) float v8f;

#define LAYERS  8
#define BATCH   64
#define DMODEL  2048
#define DINNER  4096
#define DSTATE  128
#define DTRANK  128

#define KT   128          // K-tile staged in LDS
#define KTP  (KT + 4)     // row pad (floats) to spread LDS banks

__device__ __forceinline__ float silu_f(float x) { return x / (1.0f + expf(-x)); }

enum { EP_NONE = 0, EP_BIAS = 1, EP_BIAS_SILU = 2, EP_BIAS_SOFTPLUS = 3, EP_ADD = 4 };

// ---------------------------------------------------------------------------
// fp32 WMMA GEMM:  C[64 x N] = epilogue(A[64 x K] @ W[K x N])
// One wave32 per 64x16 output strip (4 M-tiles fused -> 4 v8f accumulators,
// B fragment reused across 4 v_wmma_f32_16x16x4_f32). The 64xKT A tile is
// staged cooperatively in LDS (shared by all 4 waves of the block) and read
// back as ds_load_b64 fragments; W streams from global with prefetch.
// Requires K % KT == 0 and lda % 4 == 0 (true for all call sites).
// ---------------------------------------------------------------------------
template <int MODE>
__global__ __launch_bounds__(128) void k_gemm_wmma(
    const float* __restrict__ A, int lda, const float* __restrict__ W,
    const float* __restrict__ bias, const float* __restrict__ addsrc,
    float* __restrict__ C, int N, int K) {
  __shared__ float As[64 * KTP];  // 33 KB

  const int lane = threadIdx.x & 31;
  const int wave = threadIdx.x >> 5;
  const int n0 = (blockIdx.x * 4 + wave) * 16;
  const int hv = lane >> 4;   // half-wave select
  const int lr = lane & 15;

  const float* Wcol = W + n0 + lr;

  v8f acc[4];
#pragma unroll
  for (int mt = 0; mt < 4; ++mt) acc[mt] = (v8f){0.f,0.f,0.f,0.f,0.f,0.f,0.f,0.f};

  for (int kt = 0; kt < K; kt += KT) {
    __syncthreads();
    // cooperative stage: 64 x KT floats = 2048 float4, 128 threads x 16
    for (int i = threadIdx.x; i < 64 * (KT / 4); i += 128) {
      const int row = i >> 5;        // / (KT/4)
      const int c4 = i & 31;
      const float4 v = *(const float4*)(A + (size_t)row * lda + kt + c4 * 4);
      *(float4*)(&As[row * KTP + c4 * 4]) = v;
    }
    __syncthreads();

#pragma unroll 4
    for (int k = 0; k < KT; k += 4) {
      const int kk = k + 2 * hv;
      v2f bb;
      bb.x = Wcol[(size_t)(kt + kk) * N];
      bb.y = Wcol[(size_t)(kt + kk + 1) * N];
      __builtin_prefetch(Wcol + (size_t)(kt + kk + KT) * N, 0, 1);  // global_prefetch_b8
#pragma unroll
      for (int mt = 0; mt < 4; ++mt) {
        v2f a = *(const v2f*)(&As[(mt * 16 + lr) * KTP + kk]);
        acc[mt] = __builtin_amdgcn_wmma_f32_16x16x4_f32(
            false, a, false, bb, (short)0, acc[mt], false, false);
      }
    }
  }

  const int col = n0 + lr;
  float bv = 0.f;
  if (MODE == EP_BIAS || MODE == EP_BIAS_SILU || MODE == EP_BIAS_SOFTPLUS)
    bv = bias[col];
#pragma unroll
  for (int mt = 0; mt < 4; ++mt) {
#pragma unroll
    for (int r = 0; r < 8; ++r) {
      const int row = mt * 16 + r + 8 * hv;
      float v = acc[mt][r];
      if (MODE == EP_BIAS) v += bv;
      if (MODE == EP_BIAS_SILU) { v += bv; v = silu_f(v); }
      if (MODE == EP_BIAS_SOFTPLUS) { v += bv; v = (v > 20.f) ? v : log1pf(expf(v)); }
      if (MODE == EP_ADD) v += addsrc[(size_t)row * N + col];
      C[(size_t)row * N + col] = v;
    }
  }
}

// ---------------------------------------------------------------------------
// t1 = silu(action(64x14) @ ae_w1(14x2048) + ae_b1)   (K=14: plain VALU kernel)
// ---------------------------------------------------------------------------
__global__ __launch_bounds__(256) void k_act_embed(
    const float* __restrict__ action, const float* __restrict__ w1,
    const float* __restrict__ b1, float* __restrict__ t1) {
  const int idx = blockIdx.x * 256 + threadIdx.x;  // b*2048 + n
  const int b = idx >> 11, n = idx & 2047;
  float acc = b1[n];
#pragma unroll
  for (int k = 0; k < 14; ++k) acc += action[b * 14 + k] * w1[k * 2048 + n];
  t1[idx] = silu_f(acc);
}

// ---------------------------------------------------------------------------
// hn[b, 0:8192] = LN(concat(z_full, a_emb, noise)) * g + bvec ; one block/row
// ---------------------------------------------------------------------------
__device__ __forceinline__ float cat_val(const float* z, const float* a,
                                         const float* n, int b, int j) {
  if (j < 4096) return z[b * 4096 + j];
  if (j < 6144) return a[b * 2048 + (j - 4096)];
  return n[b * 2048 + (j - 6144)];
}

__global__ __launch_bounds__(256) void k_concat_ln(
    const float* __restrict__ zf, const float* __restrict__ ae,
    const float* __restrict__ ne, const float* __restrict__ g,
    const float* __restrict__ bvec, float* __restrict__ out) {
  __shared__ float red[256];
  const int b = blockIdx.x;
  float s = 0.f, ss = 0.f;
  for (int j = threadIdx.x; j < 8192; j += 256) {
    const float v = cat_val(zf, ae, ne, b, j);
    s += v; ss += v * v;
  }
  red[threadIdx.x] = s; __syncthreads();
  for (int o = 128; o; o >>= 1) { if (threadIdx.x < o) red[threadIdx.x] += red[threadIdx.x + o]; __syncthreads(); }
  const float mean = red[0] * (1.f / 8192.f); __syncthreads();
  red[threadIdx.x] = ss; __syncthreads();
  for (int o = 128; o; o >>= 1) { if (threadIdx.x < o) red[threadIdx.x] += red[threadIdx.x + o]; __syncthreads(); }
  const float var = red[0] * (1.f / 8192.f) - mean * mean;
  const float inv = rsqrtf(var + 1e-5f);
  for (int j = threadIdx.x; j < 8192; j += 256) {
    const float v = cat_val(zf, ae, ne, b, j);
    out[(size_t)b * 8192 + j] = (v - mean) * inv * g[j] + bvec[j];
  }
}

// Generic row LayerNorm (used for final 2048-wide LN)
__global__ __launch_bounds__(256) void k_layernorm(
    const float* __restrict__ x, const float* __restrict__ g,
    const float* __restrict__ bvec, float* __restrict__ out, int width) {
  __shared__ float red[256];
  const int b = blockIdx.x;
  const float* xr = x + (size_t)b * width;
  float s = 0.f, ss = 0.f;
  for (int j = threadIdx.x; j < width; j += 256) { const float v = xr[j]; s += v; ss += v * v; }
  red[threadIdx.x] = s; __syncthreads();
  for (int o = 128; o; o >>= 1) { if (threadIdx.x < o) red[threadIdx.x] += red[threadIdx.x + o]; __syncthreads(); }
  const float mean = red[0] / (float)width; __syncthreads();
  red[threadIdx.x] = ss; __syncthreads();
  for (int o = 128; o; o >>= 1) { if (threadIdx.x < o) red[threadIdx.x] += red[threadIdx.x + o]; __syncthreads(); }
  const float var = red[0] / (float)width - mean * mean;
  const float inv = rsqrtf(var + 1e-5f);
  for (int j = threadIdx.x; j < width; j += 256)
    out[(size_t)b * width + j] = (xr[j] - mean) * inv * g[j] + bvec[j];
}

// ---------------------------------------------------------------------------
// Conv-state shift + depthwise conv + silu. One thread per (b,d). float4 IO.
// Writes new conv state straight into d_out's conv region.
// ---------------------------------------------------------------------------
__global__ __launch_bounds__(256) void k_conv(
    const float* __restrict__ xz, const float* __restrict__ conv_in,
    const float* __restrict__ cw, const float* __restrict__ cb,
    float* __restrict__ conv_out, float* __restrict__ xc) {
  const int idx = blockIdx.x * 256 + threadIdx.x;  // b*4096 + d
  const int b = idx >> 12, d = idx & 4095;
  const float x = xz[(size_t)b * 8192 + d];
  const float4 cs = *(const float4*)(conv_in + (size_t)idx * 4);
  const float4 w = *(const float4*)(cw + (size_t)d * 4);
  const float4 cn = make_float4(cs.y, cs.z, cs.w, x);
  *(float4*)(conv_out + (size_t)idx * 4) = cn;
  const float acc = cn.x * w.x + cn.y * w.y + cn.z * w.z + cn.w * w.w + cb[d];
  xc[idx] = silu_f(acc);
}

// ---------------------------------------------------------------------------
// SSM state update — the HBM-bound hot loop (2 x 1.07 GB of state traffic).
// One wave32 per (b,d); each lane owns s = 4*lane..4*lane+3, so every stream
// is a fully-coalesced 512B b128 transaction per wave. y reduced via
// wave32 shfl_xor; gate with silu(z); new_ssm written straight to d_out.
// ---------------------------------------------------------------------------
__global__ __launch_bounds__(256) void k_ssm(
    const float* __restrict__ ss_in, const float* __restrict__ alog,
    const float* __restrict__ dt, const float* __restrict__ xc,
    const float* __restrict__ dbc, const float* __restrict__ Dp,
    const float* __restrict__ xz, float* __restrict__ ss_out,
    float* __restrict__ ybuf) {
  const int lane = threadIdx.x & 31;
  const int gid = blockIdx.x * 8 + (threadIdx.x >> 5);  // b*4096 + d
  const int b = gid >> 12, d = gid & 4095;
  const float dtv = dt[gid];
  const float xcv = xc[gid];
  const float dx = dtv * xcv;
  const int so = lane * 4;

  const float4 ss4 = *(const float4*)(ss_in + (size_t)gid * 128 + so);
  const float4 al4 = *(const float4*)(alog + (size_t)d * 128 + so);
  const float4 b4  = *(const float4*)(dbc + (size_t)b * 384 + DTRANK + so);
  const float4 c4  = *(const float4*)(dbc + (size_t)b * 384 + DTRANK + DSTATE + so);

  float4 sn;
  sn.x = ss4.x * expf(dtv * -expf(al4.x)) + dx * b4.x;
  sn.y = ss4.y * expf(dtv * -expf(al4.y)) + dx * b4.y;
  sn.z = ss4.z * expf(dtv * -expf(al4.z)) + dx * b4.z;
  sn.w = ss4.w * expf(dtv * -expf(al4.w)) + dx * b4.w;
  *(float4*)(ss_out + (size_t)gid * 128 + so) = sn;

  float y = sn.x * c4.x + sn.y * c4.y + sn.z * c4.z + sn.w * c4.w;
  for (int o = 16; o >= 1; o >>= 1) y += __shfl_xor(y, o, 32);
  if (lane == 0) {
    const float z = xz[(size_t)b * 8192 + 4096 + d];
    ybuf[gid] = (y + Dp[d] * xcv) * silu_f(z);
  }
}

// ---------------------------------------------------------------------------
extern "C" void kernel_launch(void* const* d_in, const int* in_sizes, int n_in,
                              void* d_out, int out_size, void* d_ws,
                              size_t ws_size, hipStream_t stream) {
  const float* z_full     = (const float*)d_in[0];
  const float* action     = (const float*)d_in[1];
  const float* noise      = (const float*)d_in[2];
  const float* ssm_states = (const float*)d_in[3];
  const float* conv_states= (const float*)d_in[4];
  const float* ae_w1      = (const float*)d_in[5];
  const float* ae_b1      = (const float*)d_in[6];
  const float* ae_w2      = (const float*)d_in[7];
  const float* ae_b2      = (const float*)d_in[8];
  const float* ip_g       = (const float*)d_in[9];
  const float* ip_b       = (const float*)d_in[10];
  const float* ip_w       = (const float*)d_in[11];
  const float* ip_bias    = (const float*)d_in[12];
  const float* in_proj_w  = (const float*)d_in[13];
  const float* conv_w     = (const float*)d_in[14];
  const float* conv_b     = (const float*)d_in[15];
  const float* x_proj_w   = (const float*)d_in[16];
  const float* dt_w       = (const float*)d_in[17];
  const float* dt_b       = (const float*)d_in[18];
  const float* A_log      = (const float*)d_in[19];
  const float* Dp         = (const float*)d_in[20];
  const float* out_w      = (const float*)d_in[21];
  const float* op_g       = (const float*)d_in[22];
  const float* op_b       = (const float*)d_in[23];
  const float* op_w       = (const float*)d_in[24];
  const float* op_bias    = (const float*)d_in[25];

  float* out      = (float*)d_out;
  float* out_dz   = out;                                        // 64*2048
  float* out_ssm  = out + (size_t)BATCH * DMODEL;               // 8*64*4096*128
  float* out_conv = out_ssm + (size_t)LAYERS * BATCH * DINNER * DSTATE;

  float* ws   = (float*)d_ws;
  float* t1   = ws;              // 64*2048
  float* aemb = t1 + 131072;     // 64*2048
  float* hn   = aemb + 131072;   // 64*8192
  float* h    = hn + 524288;     // 64*2048
  float* xz   = h + 131072;      // 64*8192
  float* xc   = xz + 524288;     // 64*4096
  float* dbc  = xc + 262144;     // 64*384
  float* dt   = dbc + 24576;     // 64*4096
  float* ybuf = dt + 262144;     // 64*4096
  float* hn2  = ybuf + 262144;   // 64*2048

  // --- input stem ---
  k_act_embed<<<512, 256, 0, stream>>>(action, ae_w1, ae_b1, t1);
  k_gemm_wmma<EP_BIAS><<<dim3(2048 / 64), 128, 0, stream>>>(
      t1, 2048, ae_w2, ae_b2, nullptr, aemb, 2048, 2048);
  k_concat_ln<<<BATCH, 256, 0, stream>>>(z_full, aemb, noise, ip_g, ip_b, hn);
  k_gemm_wmma<EP_BIAS_SILU><<<dim3(2048 / 64), 128, 0, stream>>>(
      hn, 8192, ip_w, ip_bias, nullptr, h, 2048, 8192);

  // --- 8 sequential Mamba layers ---
  for (int l = 0; l < LAYERS; ++l) {
    const float* ipw  = in_proj_w + (size_t)l * DMODEL * 2 * DINNER;
    const float* cin  = conv_states + (size_t)l * BATCH * DINNER * 4;
    float*       cout = out_conv + (size_t)l * BATCH * DINNER * 4;
    const float* cw   = conv_w + (size_t)l * DINNER * 4;
    const float* cb   = conv_b + (size_t)l * DINNER;
    const float* xpw  = x_proj_w + (size_t)l * DINNER * 384;
    const float* dtw  = dt_w + (size_t)l * DTRANK * DINNER;
    const float* dtb  = dt_b + (size_t)l * DINNER;
    const float* al   = A_log + (size_t)l * DINNER * DSTATE;
    const float* dp   = Dp + (size_t)l * DINNER;
    const float* ow   = out_w + (size_t)l * DINNER * DMODEL;
    const float* ssin = ssm_states + (size_t)l * BATCH * DINNER * DSTATE;
    float*       ssot = out_ssm + (size_t)l * BATCH * DINNER * DSTATE;

    k_gemm_wmma<EP_NONE><<<dim3(8192 / 64), 128, 0, stream>>>(
        h, 2048, ipw, nullptr, nullptr, xz, 8192, 2048);
    k_conv<<<(BATCH * DINNER) / 256, 256, 0, stream>>>(xz, cin, cw, cb, cout, xc);
    k_gemm_wmma<EP_NONE><<<dim3(384 / 64), 128, 0, stream>>>(
        xc, 4096, xpw, nullptr, nullptr, dbc, 384, 4096);
    k_gemm_wmma<EP_BIAS_SOFTPLUS><<<dim3(4096 / 64), 128, 0, stream>>>(
        dbc, 384, dtw, dtb, nullptr, dt, 4096, 128);
    k_ssm<<<(BATCH * DINNER) / 8, 256, 0, stream>>>(
        ssin, al, dt, xc, dbc, dp, xz, ssot, ybuf);
    k_gemm_wmma<EP_ADD><<<dim3(2048 / 64), 128, 0, stream>>>(
        ybuf, 4096, ow, nullptr, h, h, 2048, 4096);
  }

  // --- output head ---
  k_layernorm<<<BATCH, 256, 0, stream>>>(h, op_g, op_b, hn2, 2048);
  k_gemm_wmma<EP_BIAS><<<dim3(2048 / 64), 128, 0, stream>>>(
      hn2, 2048, op_w, op_bias, nullptr, out_dz, 2048, 2048);
}